// ModelNew_14061722927819
// MI455X (gfx1250) — compile-verified
//
#include <hip/hip_runtime.h>
#include <hip/hip_bf16.h>
#include <math.h>

typedef float v2f  __attribute__((ext_vector_type(2)));
typedef float v8f  __attribute__((ext_vector_type(8)));
typedef unsigned int u32x4 __attribute__((ext_vector_type(4)));
typedef int   i32x4 __attribute__((ext_vector_type(4)));
typedef int   i32x8 __attribute__((ext_vector_type(8)));

#define B_N    128
#define CI     8
#define HW     128
#define CO     64
#define HO     126        // conv output H=W
#define PO     31         // pooled output H=W
#define KTOT   72         // Ci * 3 * 3
#define NSTEP  18         // KTOT / 4 (K=4 per f32 WMMA)
#define NGROUP 16
#define GSIZE  4          // channels per group

// ---------------------------------------------------------------------------
// TDM: DMA an [CI][nrows][HW] f32 slab (rows starting at gptr) into LDS.
// D# per CDNA5 ISA:
//   group0: count=1 | lds_addr | global_addr[56:0] | type=2
//   group1: data_size=4B; tensor_dim0=HW, tensor_dim1=nrows;
//           tile 128 x nrows x 8; dim0_stride=HW, dim1_stride=HW*HW
//   group2: tensor_dim2=CI (3D tensor); group3: zero
// LDS written contiguously X,Y,Z -> layout [ci][row][col].
// This toolchain declares the 6-arg builtin:
//   (u32x4 g0, i32x8 g1, i32x4 g2, i32x4 g3, i32x8 g4, i32 cpol)
// g4 zero-filled (probe-verified call shape).
// Issue from wave 0 only; completion via s_wait_tensorcnt + barrier.
// ---------------------------------------------------------------------------
__device__ __forceinline__ void tdm_load_slab(unsigned lds_off,
                                              const float* gptr, int nrows) {
  unsigned long long ga = (unsigned long long)(uintptr_t)gptr;
  u32x4 g0;
  g0[0] = 1u;                                   // count=1 (valid), user mode
  g0[1] = lds_off;                              // lds_addr (bytes)
  g0[2] = (unsigned)(ga & 0xFFFFFFFFu);         // global_addr[31:0]
  g0[3] = (unsigned)((ga >> 32) & 0x01FFFFFFu)  // global_addr[56:32]
          | 0x80000000u;                        // type=2 (bits 127:126 = 10b)
  i32x8 g1;
  g1[0] = 0x00020000;                           // data_size=2 (4B), mask=0
  g1[1] = (int)(((unsigned)HW & 0xFFFFu) << 16);      // tensor_dim0 lo16@b48
  g1[2] = (int)(((unsigned)nrows & 0xFFFFu) << 16);   // dim0 hi16=0 | dim1 lo16
  g1[3] = (int)(((unsigned)HW & 0xFFFFu) << 16);      // dim1 hi16=0 | tile_dim0
  g1[4] = (int)(((unsigned)nrows) | ((unsigned)CI << 16)); // tile_dim1|tile_dim2
  g1[5] = HW;                                   // tensor_dim0_stride lo32
  g1[6] = (int)(((unsigned)(HW * HW) & 0xFFFFu) << 16); // d0s hi16=0|d1s lo16
  g1[7] = (int)((unsigned)(HW * HW) >> 16);     // tensor_dim1_stride hi32
  i32x4 g2;
  g2[0] = CI;                                   // tensor_dim2
  g2[1] = 0; g2[2] = 0; g2[3] = 0;              // dim3 / dim2_stride / tile_dim3
  i32x4 g3; g3[0] = 0; g3[1] = 0; g3[2] = 0; g3[3] = 0;
  i32x8 g4;
  g4[0] = 0; g4[1] = 0; g4[2] = 0; g4[3] = 0;
  g4[4] = 0; g4[5] = 0; g4[6] = 0; g4[7] = 0;
  __builtin_amdgcn_tensor_load_to_lds(g0, g1, g2, g3, g4, 0);
}

// ---------------------------------------------------------------------------
// Kernel 1: conv (implicit GEMM, V_WMMA_F32_16X16X4_F32) + per-(b,group)
// sum/sumsq. One WG per (batch, 16-channel m-tile). x slab for each conv row
// (3 rows x 8 ci x 128 cols = 12 KB) is TDM-DMA'd into double-buffered LDS;
// all 8 waves (one 16-col group each) assemble B operands from LDS.
// ---------------------------------------------------------------------------
__global__ __launch_bounds__(256) void conv_stats_kernel(
    const float* __restrict__ x, const float* __restrict__ w,
    const float* __restrict__ cbias, float* __restrict__ stats)
{
  __shared__ float xs[2][CI * 3 * HW];   // 2 x 12 KB slabs
  __shared__ float red[256 * 8];         // 8 KB reduction buffer

  const int tid  = threadIdx.x;
  const int lane = tid & 31;
  const int wave = tid >> 5;
  const int half = lane >> 4;
  const int l15  = lane & 15;
  const int mt   = blockIdx.x & 3;
  const int b    = blockIdx.x >> 2;
  const int co_base = mt * 16;
  const int koff = half * 2;             // A/B K sub-offset per lane half

  const float* xb = x + (size_t)b * CI * HW * HW;

  // A (weights), ISA 16x4 f32 layout: row M=l15; lanes<16 K=4s..4s+1,
  // lanes>=16 K=4s+2..4s+3.
  v2f A[NSTEP];
  const float* wrow = w + (co_base + l15) * KTOT + koff;
#pragma unroll
  for (int s = 0; s < NSTEP; ++s) {
    A[s].x = wrow[4 * s + 0];
    A[s].y = wrow[4 * s + 1];
  }
  // LDS-relative B offsets: slab layout [ci][3][HW]
  int po0[NSTEP], po1[NSTEP];
#pragma unroll
  for (int s = 0; s < NSTEP; ++s) {
    int k0 = 4 * s + koff, k1 = k0 + 1;
    int ci0 = k0 / 9, r0 = k0 % 9;
    int ci1 = k1 / 9, r1 = k1 % 9;
    po0[s] = ci0 * (3 * HW) + (r0 / 3) * HW + (r0 % 3);
    po1[s] = ci1 * (3 * HW) + (r1 / 3) * HW + (r1 % 3);
  }
  float pb[8];
#pragma unroll
  for (int r = 0; r < 8; ++r) pb[r] = cbias[co_base + r + half * 8];

  const int cg = wave;                         // each wave owns one col-group
  const int wo = cg * 16 + l15;
  const float mask = (wo < HO) ? 1.0f : 0.0f;  // cols 126/127 masked from stats
  const int woc = (wo < HO) ? wo : (HO - 1);   // clamped LDS addressing

  // Prologue: DMA slab for ho=0
  if (wave == 0)
    tdm_load_slab((unsigned)(uintptr_t)&xs[0][0], xb, 3);

  float s_lo = 0.f, q_lo = 0.f, s_hi = 0.f, q_hi = 0.f;

  for (int ho = 0; ho < HO; ++ho) {
    // Software pipeline: issue next slab, then wait current slab complete.
    if (wave == 0) {
      if (ho + 1 < HO) {
        tdm_load_slab((unsigned)(uintptr_t)&xs[(ho + 1) & 1][0],
                      xb + (size_t)(ho + 1) * HW, 3);
        __builtin_amdgcn_s_wait_tensorcnt(1);  // oldest (ho) done, next in flight
      } else {
        __builtin_amdgcn_s_wait_tensorcnt(0);
      }
    }
    __syncthreads();                           // slab[ho&1] visible to all

    const float* bp = &xs[ho & 1][woc];
    v8f c = {};
#pragma unroll
    for (int s = 0; s < NSTEP; ++s) {
      v2f Bm;
      Bm.x = bp[po0[s]];
      Bm.y = bp[po1[s]];
      c = __builtin_amdgcn_wmma_f32_16x16x4_f32(
          false, A[s], false, Bm, (short)0, c, false, false);
    }
#pragma unroll
    for (int r = 0; r < 8; ++r) {
      float y  = c[r] + pb[r];
      float ym = y * mask;
      float yq = y * ym;
      if (r < 4) { s_lo += ym; q_lo += yq; }
      else       { s_hi += ym; q_hi += yq; }
    }
    __syncthreads();                           // all reads done before overwrite
  }

  // Deterministic serial reduction over the workgroup.
  float* rp = red + tid * 8;
#pragma unroll
  for (int j = 0; j < 8; ++j) rp[j] = 0.f;
  const int gb = half * 2;   // local groups 0,1 (lanes<16) / 2,3 (lanes>=16)
  rp[gb * 2 + 0] = s_lo; rp[gb * 2 + 1] = q_lo;
  rp[gb * 2 + 2] = s_hi; rp[gb * 2 + 3] = q_hi;
  __syncthreads();
  if (tid < 8) {             // tid = g'*2 + stat
    float acc = 0.f;
    for (int i = 0; i < 256; ++i) acc += red[i * 8 + tid];
    stats[(b * NGROUP + mt * 4 + (tid >> 1)) * 2 + (tid & 1)] = acc;
  }
}

// ---------------------------------------------------------------------------
// Kernel 2: one WG per (batch, pooled row). TDM loads the 6-row x slab
// (24 KB) once; weights (18 KB) and the folded per-channel affine (ka/kb)
// are staged in LDS and reused by all 8 waves x 4 m-tiles. Each wave:
// recompute conv via WMMA for 4 conv rows, affine, 4x4 maxpool
// (rows in regs, cols via shfl_xor), clamp, store.
// ---------------------------------------------------------------------------
__global__ __launch_bounds__(256) void norm_pool_kernel(
    const float* __restrict__ x, const float* __restrict__ w,
    const float* __restrict__ cbias, const float* __restrict__ gnw,
    const float* __restrict__ gnb, const float* __restrict__ scale,
    const float* __restrict__ stats, float* __restrict__ out)
{
  __shared__ float xs[CI * 6 * HW];      // 24 KB x slab [ci][6][HW]
  __shared__ float wl[CO * KTOT];        // 18 KB weights
  __shared__ float kal[CO], kbl[CO];     // folded affine per channel

  const int tid  = threadIdx.x;
  const int lane = tid & 31;
  const int wave = tid >> 5;
  const int half = lane >> 4;
  const int l15  = lane & 15;
  const int pr   = blockIdx.x % PO;      // pooled row
  const int b    = blockIdx.x / PO;      // batch
  const int koff = half * 2;

  const float* xb = x + (size_t)b * CI * HW * HW;

  // DMA x rows 4pr..4pr+5 (conv rows + 3x3 halo), shared by every wave/m-tile.
  if (wave == 0)
    tdm_load_slab((unsigned)(uintptr_t)&xs[0], xb + (size_t)(4 * pr) * HW, 6);

  // Stage weights cooperatively while the DMA is in flight.
  for (int i = tid; i < CO * KTOT; i += 256) wl[i] = w[i];

  // Fold GN + scale + conv bias: yn = conv * ka[c] + kb[c]
  if (tid < CO) {
    const int c = tid, g = c >> 2;
    const float invN = 1.0f / (float)(GSIZE * HO * HO);
    const float sum  = stats[(b * NGROUP + g) * 2 + 0];
    const float sq   = stats[(b * NGROUP + g) * 2 + 1];
    const float mean = sum * invN;
    const float var  = sq * invN - mean * mean;
    const float rstd = rsqrtf(var + 1e-5f);
    const float aa   = rstd * gnw[c] * scale[c];
    kal[c] = aa;
    kbl[c] = (cbias[c] - mean) * aa + gnb[c] * scale[c];
  }

  if (wave == 0) __builtin_amdgcn_s_wait_tensorcnt(0);
  __syncthreads();

  // B offsets within the 6-row slab: [ci][6][HW]
  int po0[NSTEP], po1[NSTEP];
#pragma unroll
  for (int s = 0; s < NSTEP; ++s) {
    int k0 = 4 * s + koff, k1 = k0 + 1;
    int ci0 = k0 / 9, r0 = k0 % 9;
    int ci1 = k1 / 9, r1 = k1 % 9;
    po0[s] = ci0 * (6 * HW) + (r0 / 3) * HW + (r0 % 3);
    po1[s] = ci1 * (6 * HW) + (r1 / 3) * HW + (r1 % 3);
  }

  const int pcg = wave;                        // pooled-col group 0..7
  const int wo  = pcg * 16 + l15;
  const int woc = (wo < HO) ? wo : (HO - 1);   // clamp feeds only pcol 31 (masked)

  for (int mt = 0; mt < 4; ++mt) {
    const int co_base = mt * 16;
    // A from LDS
    v2f A[NSTEP];
    const float* wrow = &wl[(co_base + l15) * KTOT + koff];
#pragma unroll
    for (int s = 0; s < NSTEP; ++s) {
      A[s].x = wrow[4 * s + 0];
      A[s].y = wrow[4 * s + 1];
    }
    float ka[8], kb[8];
#pragma unroll
    for (int r = 0; r < 8; ++r) {
      const int c = co_base + r + half * 8;
      ka[r] = kal[c];
      kb[r] = kbl[c];
    }

    float mrow[8];
#pragma unroll
    for (int r = 0; r < 8; ++r) mrow[r] = -INFINITY;

    for (int rr = 0; rr < 4; ++rr) {           // 4 conv rows of the pool window
      const float* bp = &xs[rr * HW + woc];
      v8f c = {};
#pragma unroll
      for (int s = 0; s < NSTEP; ++s) {
        v2f Bm;
        Bm.x = bp[po0[s]];
        Bm.y = bp[po1[s]];
        c = __builtin_amdgcn_wmma_f32_16x16x4_f32(
            false, A[s], false, Bm, (short)0, c, false, false);
      }
#pragma unroll
      for (int r = 0; r < 8; ++r)
        mrow[r] = fmaxf(mrow[r], c[r] * ka[r] + kb[r]);  // affine before max
    }

    // Column max within 4-lane quads, then clamp to [0,1].
#pragma unroll
    for (int r = 0; r < 8; ++r) {
      float m = mrow[r];
      m = fmaxf(m, __shfl_xor(m, 1, 32));
      m = fmaxf(m, __shfl_xor(m, 2, 32));
      mrow[r] = fminf(fmaxf(m, 0.0f), 1.0f);
    }

    const int pcol = pcg * 4 + (l15 >> 2);
    if (((l15 & 3) == 0) && pcol < PO) {
#pragma unroll
      for (int r = 0; r < 8; ++r) {
        const int c = co_base + r + half * 8;
        out[(((size_t)b * CO + c) * PO + pr) * PO + pcol] = mrow[r];
      }
    }
  }
}

// ---------------------------------------------------------------------------
extern "C" void kernel_launch(void* const* d_in, const int* in_sizes, int n_in,
                              void* d_out, int out_size, void* d_ws, size_t ws_size,
                              hipStream_t stream) {
  const float* x     = (const float*)d_in[0];  // [128,8,128,128]
  const float* w     = (const float*)d_in[1];  // [64,8,3,3]
  const float* cbias = (const float*)d_in[2];  // [64]
  const float* gnw   = (const float*)d_in[3];  // [64]
  const float* gnb   = (const float*)d_in[4];  // [64]
  const float* sc    = (const float*)d_in[5];  // [64,1,1]
  float* stats = (float*)d_ws;                 // [128][16][2] = 16 KB
  float* out   = (float*)d_out;                // [128,64,31,31]

  conv_stats_kernel<<<B_N * 4, 256, 0, stream>>>(x, w, cbias, stats);
  norm_pool_kernel<<<B_N * PO, 256, 0, stream>>>(
      x, w, cbias, gnw, gnb, sc, stats, out);
}